// Linear_21809843929404
// MI455X (gfx1250) — compile-verified
//
#include <hip/hip_runtime.h>
#include <hip/hip_bf16.h>

typedef __attribute__((ext_vector_type(2))) float v2f;
typedef __attribute__((ext_vector_type(8))) float v8f;

#define NAGT 4096
#define NTILE (NAGT / 16)

// Exponent computed directly in base 2 so v_exp_f32 needs no *log2e mul:
//   10*exp((0.6-d)/0.71) = exp2(C2 - d*INVB2)
#define LOG2E 1.4426950408889634f
#define INVB2 (LOG2E / 0.71f)
#define C2 ((0.6f / 0.71f + 2.3025850929940457f) * LOG2E)

// ---------------------------------------------------------------------------
// One 16x16 pairwise tile: WMMA Gram tile + per-element social-force weight.
// DIAG instantiation adds the i==j mask; only the single 16-aligned tile that
// intersects the diagonal needs it (selected by a *scalar* branch).
// rr2e[] already includes +EPS.
// ---------------------------------------------------------------------------
template <bool DIAG>
__device__ __forceinline__ void tile_body(
    const v2f& a, float xc, float yc, bool hi, int l15, int rbase, int jt,
    const float rr2e[8], float accS[8], float accTx[8], float accTy[8]) {
  // B tile (4x16 f32): lanes 0-15 hold N=lane with K0=x, K1=y; hi half = 0.
  v2f b;
  b.x = hi ? 0.0f : xc;
  b.y = hi ? 0.0f : yc;

  v8f c = {};
  v8f dmat = __builtin_amdgcn_wmma_f32_16x16x4_f32(
      /*neg_a=*/false, a, /*neg_b=*/false, b,
      /*c_mod=*/(short)0, c, /*reuse_a=*/false, /*reuse_b=*/false);

  const float cc2 = xc * xc + yc * yc;
  const int col   = jt * 16 + l15;

#pragma unroll
  for (int r = 0; r < 8; ++r) {
    float t    = fmaxf(fmaf(-2.0f, dmat[r], rr2e[r] + cc2), 1e-8f);
    float invd = rsqrtf(t);
    float dist = t * invd;                                // sqrt(d2 + eps)
    float w    = __builtin_amdgcn_exp2f(C2 - dist * INVB2) * invd;
    if (DIAG) {
      if (rbase + r == col) w = 0.0f;                     // i == j
    }
    accS[r] += w;
    accTx[r] = fmaf(w, xc, accTx[r]);
    accTy[r] = fmaf(w, yc, accTy[r]);
  }
}

// ---------------------------------------------------------------------------
// Kernel 1: pairwise social forces.
// Each wave owns 16 rows; loops over 256 column tiles with register
// double-buffering of the column positions. rf_i = p_i*S_i - T_i.
// ---------------------------------------------------------------------------
__global__ __launch_bounds__(256) void forces_kernel(
    const float* __restrict__ state,
    float* __restrict__ Sg, float* __restrict__ Txg, float* __restrict__ Tyg) {
  const int lane   = threadIdx.x & 31;
  const int wave   = threadIdx.x >> 5;
  const int rowTop = blockIdx.x * 128 + wave * 16;  // 16 rows per wave
  const bool hi    = lane >= 16;
  const int  l15   = lane & 15;
  // rowTop>>4 is wave-uniform; force it into an SGPR so the diag-tile
  // selection below is a scalar s_cmp/s_cbranch.
  const int diagTile = __builtin_amdgcn_readfirstlane(rowTop >> 4);

  // A tile (16x4 f32): lanes 0-15 hold M=lane with K0=x, K1=y;
  // lanes 16-31 hold K2,K3 which are zero (K padded 2->4).
  v2f a;
  if (!hi) {
    a.x = state[(rowTop + lane) * 4 + 0];
    a.y = state[(rowTop + lane) * 4 + 1];
  } else {
    a.x = 0.0f; a.y = 0.0f;
  }

  // This lane's 8 rows of the D tile: rows rbase..rbase+7. EPS folded in.
  const int rbase = rowTop + (hi ? 8 : 0);
  float rr2e[8];
#pragma unroll
  for (int r = 0; r < 8; ++r) {
    float xr = state[(rbase + r) * 4 + 0];
    float yr = state[(rbase + r) * 4 + 1];
    rr2e[r] = (xr * xr + yr * yr) + 1e-8f;
  }

  float accS[8], accTx[8], accTy[8];
#pragma unroll
  for (int r = 0; r < 8; ++r) { accS[r] = 0.0f; accTx[r] = 0.0f; accTy[r] = 0.0f; }

  // Register double-buffer of column positions: load tile jt+1 while
  // computing tile jt, hiding L1/L2 latency behind the tile body.
  float xc = state[l15 * 4 + 0];
  float yc = state[l15 * 4 + 1];
  for (int jt = 0; jt < NTILE; ++jt) {
    const int nt = (jt + 1 < NTILE) ? (jt + 1) : (NTILE - 1);  // clamp, no branch
    const int nc = nt * 16 + l15;
    const float xn = state[nc * 4 + 0];
    const float yn = state[nc * 4 + 1];

    if (jt == diagTile) {   // scalar branch (diagTile is in an SGPR)
      tile_body<true>(a, xc, yc, hi, l15, rbase, jt, rr2e, accS, accTx, accTy);
    } else {
      tile_body<false>(a, xc, yc, hi, l15, rbase, jt, rr2e, accS, accTx, accTy);
    }
    xc = xn; yc = yn;
  }

  // Reduce over the 16 columns held across each 16-lane half (wave32).
#pragma unroll
  for (int r = 0; r < 8; ++r) {
#pragma unroll
    for (int m = 8; m >= 1; m >>= 1) {
      accS[r]  += __shfl_xor(accS[r],  m, 16);
      accTx[r] += __shfl_xor(accTx[r], m, 16);
      accTy[r] += __shfl_xor(accTy[r], m, 16);
    }
  }
  if (l15 == 0) {
#pragma unroll
    for (int r = 0; r < 8; ++r) {
      Sg[rbase + r]  = accS[r];
      Txg[rbase + r] = accTx[r];
      Tyg[rbase + r] = accTy[r];
    }
  }
}

// ---------------------------------------------------------------------------
// Kernel 2: per-agent goal force + Euler integration + speed clamp.
// ---------------------------------------------------------------------------
__global__ __launch_bounds__(256) void update_kernel(
    const float* __restrict__ state, const float* __restrict__ goals,
    const float* __restrict__ Sg, const float* __restrict__ Txg,
    const float* __restrict__ Tyg, float* __restrict__ out) {
  const int i = blockIdx.x * blockDim.x + threadIdx.x;
  if (i >= NAGT) return;
  float px = state[i * 4 + 0], py = state[i * 4 + 1];
  float vx = state[i * 4 + 2], vy = state[i * 4 + 3];
  float tx = goals[i * 2 + 0] - px, ty = goals[i * 2 + 1] - py;
  float dg = sqrtf(tx * tx + ty * ty + 1e-8f);
  float spd = (i == 0) ? 1.0f : 1.4f;            // robot vs pedestrians
  float inv = spd / dg;
  float afx = 2.0f * (tx * inv - vx);            // K*(v_des - vel)
  float afy = 2.0f * (ty * inv - vy);
  float s = Sg[i];
  float Fx = (px * s - Txg[i]) + afx;
  float Fy = (py * s - Tyg[i]) + afy;
  const float sc = 0.1f / 60.0f;                 // DT / PED_MASS
  float vnx = vx + Fx * sc;
  float vny = vy + Fy * sc;
  float sp = sqrtf(vnx * vnx + vny * vny + 1e-8f);
  float cl = fminf(1.0f, 1.4f / sp);
  vnx *= cl; vny *= cl;
  out[i * 4 + 0] = px + vnx * 0.1f;
  out[i * 4 + 1] = py + vny * 0.1f;
  out[i * 4 + 2] = vnx;
  out[i * 4 + 3] = vny;
}

// ---------------------------------------------------------------------------
// Kernel 3: cost update (needs out[0] -> stream-ordered after update_kernel).
// ---------------------------------------------------------------------------
__global__ __launch_bounds__(256) void cost_kernel(
    const float* __restrict__ out, const float* __restrict__ cost_in,
    const float* __restrict__ robot_init, const float* __restrict__ goal,
    float* __restrict__ cost_out) {
  const int i = blockIdx.x * blockDim.x + threadIdx.x;
  if (i >= NAGT) return;
  float rpx = out[0], rpy = out[1];
  float gx = goal[0], gy = goal[1];
  float rix = robot_init[0], riy = robot_init[1];
  float pg = sqrtf((rix - gx) * (rix - gx) + (riy - gy) * (riy - gy) + 1e-8f) -
             sqrtf((rpx - gx) * (rpx - gx) + (rpy - gy) * (rpy - gy) + 1e-8f);
  float dx = out[i * 4 + 0] - rpx, dy = out[i * 4 + 1] - rpy;
  float dist = sqrtf(dx * dx + dy * dy + 1e-8f);
  // 2*exp(-d/1.5) = 2*exp2(-d * log2e/1.5)
  float temp = 5.0f * pg +
               2.0f * __builtin_amdgcn_exp2f(-dist * (LOG2E / 1.5f));
  cost_out[i] = cost_in[i] + temp;
}

extern "C" void kernel_launch(void* const* d_in, const int* in_sizes, int n_in,
                              void* d_out, int out_size, void* d_ws, size_t ws_size,
                              hipStream_t stream) {
  (void)in_sizes; (void)n_in; (void)out_size; (void)ws_size;
  const float* state      = (const float*)d_in[0];  // (N,4)
  const float* cost       = (const float*)d_in[1];  // (N,1)
  const float* goals      = (const float*)d_in[2];  // (N,2)
  const float* robot_init = (const float*)d_in[3];  // (2,)
  const float* goal       = (const float*)d_in[4];  // (2,)

  float* out      = (float*)d_out;           // (N,4) flat
  float* cost_out = out + NAGT * 4;          // (N,1) flat, concatenated

  float* Sg  = (float*)d_ws;                 // N floats
  float* Txg = Sg + NAGT;                    // N floats
  float* Tyg = Txg + NAGT;                   // N floats

  // 32 blocks * 8 waves * 16 rows = 4096 rows
  forces_kernel<<<NAGT / 128, 256, 0, stream>>>(state, Sg, Txg, Tyg);
  update_kernel<<<NAGT / 256, 256, 0, stream>>>(state, goals, Sg, Txg, Tyg, out);
  cost_kernel<<<NAGT / 256, 256, 0, stream>>>(out, cost, robot_init, goal, cost_out);
}